// PlackettLuceModel_74955769250738
// MI455X (gfx1250) — compile-verified
//
#include <hip/hip_runtime.h>
#include <math.h>

typedef __attribute__((ext_vector_type(2))) float v2f;
typedef __attribute__((ext_vector_type(8))) float v8f;

#define NPTS 4096   // N: list length
#define DDIM 64     // D: feature dim

// ---------------------------------------------------------------------------
// Kernel 1: scores[row] = dot(X[row, 0:64], W) + bias via V_WMMA_F32_16X16X4_F32
// Each wave (32 lanes) computes 16 consecutive rows: 16 accumulating WMMAs
// over K = 64 in chunks of 4. B-matrix = W broadcast across all 16 columns,
// so every output column equals the score vector; lanes 0/16 (N==0) store it.
// A-layout (16x4 f32): VGPR0/1 = K0,K1 on lanes 0-15; K2,K3 on lanes 16-31,
// M = lane%16 -> one aligned float2 load per lane per K-chunk (no waste).
// ---------------------------------------------------------------------------
__global__ __launch_bounds__(256) void pl_scores_wmma(
    const float* __restrict__ X, const float* __restrict__ W,
    const float* __restrict__ bias, float* __restrict__ scores)
{
  const int lane    = threadIdx.x & 31;
  const int wave    = threadIdx.x >> 5;
  const int rowBase = (blockIdx.x * 8 + wave) * 16;   // 8 waves/block * 16 rows
  const int half    = lane >> 4;                      // 0: lanes 0-15, 1: 16-31
  const int m       = lane & 15;

  const float* xrow = X + (size_t)(rowBase + m) * DDIM + half * 2;

  v8f acc = {0.f, 0.f, 0.f, 0.f, 0.f, 0.f, 0.f, 0.f};
#pragma unroll
  for (int kk = 0; kk < DDIM / 4; ++kk) {
    v2f a = *(const v2f*)(xrow + kk * 4);                 // A: X tile chunk
    v2f w = *(const v2f*)(W + kk * 4 + half * 2);         // B: W broadcast
    acc = __builtin_amdgcn_wmma_f32_16x16x4_f32(
        /*neg_a=*/false, a, /*neg_b=*/false, w,
        /*c_mod=*/(short)0, acc, /*reuse_a=*/false, /*reuse_b=*/false);
  }

  const float bb = bias[0];
  // D layout: VGPR r -> M=r (lanes 0-15), M=r+8 (lanes 16-31), N = lane%16.
  // Take column N==0: lane 0 holds rows 0..7, lane 16 holds rows 8..15.
  if ((lane & 15) == 0) {
    float* dst = scores + rowBase + half * 8;
    float4 s0 = make_float4(acc[0] + bb, acc[1] + bb, acc[2] + bb, acc[3] + bb);
    float4 s1 = make_float4(acc[4] + bb, acc[5] + bb, acc[6] + bb, acc[7] + bb);
    *(float4*)(dst)     = s0;
    *(float4*)(dst + 4) = s1;
  }
}

// ---------------------------------------------------------------------------
// Kernel 2: per batch row b (one block):
//   rs[i]    = scores[b, rankings[b, i]]                 (gather into LDS)
//   denom[i] = logsumexp(rs[i:])                          (suffix scan)
//   partial  = sum_i (rs[i] - denom[i]);  out += -partial / B
// Suffix logsumexp as (M, S) pairs: lse = M + log(S).
// 3-level scan: 16-elem serial per thread -> serial 256-segment suffix ->
// second per-thread pass with carry folded in.
// ---------------------------------------------------------------------------
__device__ __forceinline__ void lse_fold(float& M, float& S, float x)
{
  if (x > M) { S = S * __expf(M - x) + 1.f; M = x; }
  else       { S += __expf(x - M); }
}

__global__ __launch_bounds__(256) void pl_nll(
    const float* __restrict__ scores, const int* __restrict__ rankings,
    float* __restrict__ out, float invB)
{
  const int b   = blockIdx.x;
  const int tid = threadIdx.x;

  __shared__ float rs[NPTS];      // 16 KB
  __shared__ float eM[256], eS[256];
  __shared__ float red[256];

  const float* srow = scores   + (size_t)b * NPTS;
  const int*   irow = rankings + (size_t)b * NPTS;

  for (int i = tid; i < NPTS; i += 256)
    rs[i] = srow[irow[i]];
  __syncthreads();

  // Level 1: per-thread segment (16 consecutive elems) reverse-scan totals.
  const int base = tid * 16;
  float M = -INFINITY, S = 0.f;
#pragma unroll
  for (int j = 15; j >= 0; --j) lse_fold(M, S, rs[base + j]);
  eM[tid] = M; eS[tid] = S;
  __syncthreads();

  // Level 2: exclusive suffix combine over 256 segment totals (serial, cheap).
  if (tid == 0) {
    float m = -INFINITY, s = 0.f;
    for (int t = 255; t >= 0; --t) {
      float tm = eM[t], ts = eS[t];
      eM[t] = m; eS[t] = s;                 // carry = segments t+1 .. 255
      if (tm > m) { s = s * __expf(m - tm) + ts; m = tm; }
      else        { s += ts * __expf(tm - m); }
    }
  }
  __syncthreads();

  // Level 3: second pass with carry; emit per-element denom and accumulate.
  float local = 0.f;
  M = eM[tid]; S = eS[tid];
#pragma unroll
  for (int j = 15; j >= 0; --j) {
    float x = rs[base + j];
    lse_fold(M, S, x);
    local += x - (M + __logf(S));
  }

  red[tid] = local;
  __syncthreads();
  for (int off = 128; off > 0; off >>= 1) {
    if (tid < off) red[tid] += red[tid + off];
    __syncthreads();
  }
  if (tid == 0) atomicAdd(out, -red[0] * invB);
}

// ---------------------------------------------------------------------------
extern "C" void kernel_launch(void* const* d_in, const int* in_sizes, int n_in,
                              void* d_out, int out_size, void* d_ws, size_t ws_size,
                              hipStream_t stream)
{
  const float* X        = (const float*)d_in[0];   // (B, N, D) f32
  const float* W        = (const float*)d_in[1];   // (D,) f32
  const float* bias     = (const float*)d_in[2];   // (1,) f32
  const int*   rankings = (const int*)  d_in[3];   // (B, N) int
  float*       out      = (float*)d_out;           // scalar f32
  float*       scores   = (float*)d_ws;            // (B, N) f32 scratch

  const int B         = in_sizes[3] / NPTS;        // 256
  const int totalRows = B * NPTS;                  // 1,048,576
  const int blocks1   = totalRows / 128;           // 256 thr = 8 waves x 16 rows

  hipMemsetAsync(d_out, 0, sizeof(float), stream);
  pl_scores_wmma<<<blocks1, 256, 0, stream>>>(X, W, bias, scores);
  pl_nll<<<B, 256, 0, stream>>>(scores, rankings, out, 1.0f / (float)B);
}